// DGCNN_83365315216148
// MI455X (gfx1250) — compile-verified
//
#include <hip/hip_runtime.h>
#include <math.h>

typedef float v2f __attribute__((ext_vector_type(2)));
typedef float v8f __attribute__((ext_vector_type(8)));

#define LRELU(v) ((v) >= 0.f ? (v) : 0.2f * (v))
#define BN_EPS 1e-5f

// ---------------------------------------------------------------------------
// Stage-1 knn: dist[i,j] = 2*xi.xj - |xi|^2 - |xj|^2 over 4096 points, C=3.
// One block per row i; top-20 via iterative argmax (stable, lowest index).
// ---------------------------------------------------------------------------
__global__ void knn_big_kernel(const float* __restrict__ x, int N, int K, int* __restrict__ idx) {
    __shared__ float dist[4096];
    __shared__ float rv[256];
    __shared__ int   ri[256];
    const int i = blockIdx.x;
    const int t = threadIdx.x;
    const float xi0 = x[i], xi1 = x[N + i], xi2 = x[2 * N + i];
    const float xxi = xi0 * xi0 + xi1 * xi1 + xi2 * xi2;
    for (int j = t; j < N; j += 256) {
        float a = x[j], b = x[N + j], c = x[2 * N + j];
        dist[j] = 2.f * (xi0 * a + xi1 * b + xi2 * c) - xxi - (a * a + b * b + c * c);
    }
    __syncthreads();
    for (int k = 0; k < K; ++k) {
        float bv = -INFINITY; int bi = N;
        for (int j = t; j < N; j += 256) {
            float v = dist[j];
            if (v > bv || (v == bv && j < bi)) { bv = v; bi = j; }
        }
        rv[t] = bv; ri[t] = bi;
        __syncthreads();
        for (int s = 128; s > 0; s >>= 1) {
            if (t < s) {
                float v2 = rv[t + s]; int i2 = ri[t + s];
                if (v2 > rv[t] || (v2 == rv[t] && i2 < ri[t])) { rv[t] = v2; ri[t] = i2; }
            }
            __syncthreads();
        }
        if (t == 0) { idx[i * K + k] = ri[0]; dist[ri[0]] = -INFINITY; }
        __syncthreads();
    }
}

// ---------------------------------------------------------------------------
// Small knn for stages 2-4: X viewed as (M, P) row-major, dist is (P, P)
// with inner dim M. One block (P threads) per row i.
// ---------------------------------------------------------------------------
__global__ void knn_small_kernel(const float* __restrict__ X, int M, int P, int K, int* __restrict__ idx) {
    __shared__ float dist[128];
    __shared__ float sq[128];
    const int i = blockIdx.x;
    const int j = threadIdx.x;      // blockDim.x == P
    float dot = 0.f, s = 0.f;
    for (int m = 0; m < M; ++m) {
        float xi = X[(size_t)m * P + i];
        float xj = X[(size_t)m * P + j];
        dot += xi * xj;
        s += xj * xj;
    }
    sq[j] = s;
    __syncthreads();
    dist[j] = 2.f * dot - sq[i] - sq[j];
    __syncthreads();
    if (j == 0) {
        for (int k = 0; k < K; ++k) {
            float best = -INFINITY; int bi = 0;
            for (int t = 0; t < P; ++t)
                if (dist[t] > best) { best = dist[t]; bi = t; }
            idx[i * K + k] = bi;
            dist[bi] = -INFINITY;
        }
    }
}

// ---------------------------------------------------------------------------
// Stage-1 conv (6->64) + BN + lrelu + max over k. Tiny; scalar kernel.
// ---------------------------------------------------------------------------
__global__ void stage1_conv_kernel(const float* __restrict__ x, const float* __restrict__ W,
                                   const float* __restrict__ g, const float* __restrict__ be,
                                   const float* __restrict__ mu, const float* __restrict__ va,
                                   const int* __restrict__ idx, float* __restrict__ out,
                                   int N, int K) {
    const int t = blockIdx.x * blockDim.x + threadIdx.x;
    const int p = t % N;
    const int o = t / N;  // o < 64
    float w[6];
    #pragma unroll
    for (int c = 0; c < 6; ++c) w[c] = W[o * 6 + c];
    const float xp0 = x[p], xp1 = x[N + p], xp2 = x[2 * N + p];
    const float ctr = w[3] * xp0 + w[4] * xp1 + w[5] * xp2;
    float hmax = -INFINITY, hmin = INFINITY;
    for (int k = 0; k < K; ++k) {
        int j = idx[p * K + k];
        float h = w[0] * (x[j] - xp0) + w[1] * (x[N + j] - xp1) + w[2] * (x[2 * N + j] - xp2) + ctr;
        hmax = fmaxf(hmax, h);
        hmin = fminf(hmin, h);
    }
    const float inv = g[o] / sqrtf(va[o] + BN_EPS);
    float a1 = (hmax - mu[o]) * inv + be[o];
    float a2 = (hmin - mu[o]) * inv + be[o];
    a1 = LRELU(a1); a2 = LRELU(a2);
    out[(size_t)o * N + p] = fmaxf(a1, a2);
}

// ---------------------------------------------------------------------------
// Build per-k effective weights directly in WMMA B-fragment order:
//   value: slot k < K : Weff[k][j][o] = sum_{c: idx[c,k]==j} W[o,c] - W[o,j]
//          slot K     : plain transpose W^T[j][o]  (also standalone via K==0)
// Fragment layout: for slot, o-tile ot=o/16, k-step s=j/4:
//   lane = (j%4)/2 * 16 + (o%16), elem = j%2  ->  contiguous float2 per lane.
//   pos  = (((slot*(O/16) + ot) * (P/4) + s) * 32 + lane) * 2 + elem
// ---------------------------------------------------------------------------
__global__ void weff_prep_kernel(float* __restrict__ weff, const float* __restrict__ W,
                                 const int* __restrict__ idx, int P, int O, int K) {
    const int t = blockIdx.x * blockDim.x + threadIdx.x;
    const int total = (K + 1) * P * O;
    if (t >= total) return;
    const int o = t % O;
    const int j = (t / O) % P;
    const int k = t / (O * P);
    float v;
    if (k == K) {
        v = W[o * P + j];
    } else {
        v = -W[o * P + j];
        for (int c = 0; c < P; ++c)
            if (idx[c * K + k] == j) v += W[o * P + c];
    }
    const int ot   = o >> 4;
    const int s    = j >> 2;
    const int rem  = j & 3;
    const int lane = ((rem >> 1) << 4) + (o & 15);
    const int e    = rem & 1;
    const size_t pos = ((((size_t)k * (O >> 4) + ot) * (P >> 2) + s) * 32 + lane) * 2 + e;
    weff[pos] = v;
}

// ---------------------------------------------------------------------------
// WMMA fp32 GEMM with fused max/min over k-slots + BN + lrelu epilogue.
//   Y[pp,o] = sum_j A[pp,j] * B[slot][j,o]
//   out[o*outLD + outOff + pp] = max(phi(max_k Y), phi(min_k Y))
// One wave per 16x16 tile (wave32, EXEC all-1s). A fragments cached in
// registers across slots; B consumed from pre-swizzled fragment buffer with
// batched b64 loads (one wait per chunk of up to 32 WMMAs) and prefetch of
// the next slot's fragments.
// ---------------------------------------------------------------------------
template <int PIN, bool ACONTIG>
__global__ void gemm_maxk_bn_kernel(const float* __restrict__ A, int rsA, int csA,
                                    int M, int O,
                                    const float* __restrict__ B, int nslots,
                                    float* __restrict__ out, int outLD, int outOff,
                                    const float* __restrict__ g, const float* __restrict__ be,
                                    const float* __restrict__ mu, const float* __restrict__ va) {
    constexpr int NS = PIN / 4;                 // wmma steps per slot
    constexpr int CH = (NS < 32) ? NS : 32;     // B-load chunk size
    const int lane = threadIdx.x & 31;
    const int wave = threadIdx.x >> 5;
    const int tilesM = M >> 4;
    const int tilesO = O >> 4;
    const int tile = blockIdx.x * (blockDim.x >> 5) + wave;
    if (tile >= tilesM * tilesO) return;
    const int ppBase = (tile % tilesM) << 4;
    const int oBase  = (tile / tilesM) << 4;
    const int ot     = oBase >> 4;

    // A fragment layout (32-bit A 16x4): lane L holds row m=L&15,
    // cols base+{0,1} for L<16 and base+{2,3} for L>=16.
    const int rowA = ppBase + (lane & 15);
    const int pairOff = (lane >> 4) << 1;  // 0 or 2
    v2f aReg[NS];
    if (ACONTIG) {
        const v2f* __restrict__ Af = (const v2f*)(A + (size_t)rowA * rsA);
        #pragma unroll
        for (int s = 0; s < NS; ++s) aReg[s] = Af[s * 2 + (pairOff >> 1)];
    } else {
        #pragma unroll
        for (int s = 0; s < NS; ++s) {
            const int c0 = s * 4 + pairOff;
            aReg[s].x = A[(size_t)rowA * rsA + (size_t)c0 * csA];
            aReg[s].y = A[(size_t)rowA * rsA + (size_t)(c0 + 1) * csA];
        }
    }

    v8f accMax, accMin;
    #pragma unroll
    for (int e = 0; e < 8; ++e) { accMax[e] = -INFINITY; accMin[e] = INFINITY; }

    const v2f* __restrict__ Bv = (const v2f*)B;
    for (int slot = 0; slot < nslots; ++slot) {
        const v2f* __restrict__ Bf = Bv + ((size_t)slot * tilesO + ot) * NS * 32;
        if (slot + 1 < nslots) {
            // keep next slot's 256*NS-byte fragment block warm (global_prefetch)
            const char* nb = (const char*)(Bv + ((size_t)(slot + 1) * tilesO + ot) * NS * 32);
            __builtin_prefetch(nb + (size_t)lane * (NS * 8), 0, 1);
        }
        v8f acc;
        #pragma unroll
        for (int e = 0; e < 8; ++e) acc[e] = 0.f;
        #pragma unroll
        for (int cb = 0; cb < NS; cb += CH) {
            v2f bReg[CH];
            #pragma unroll
            for (int c = 0; c < CH; ++c) bReg[c] = Bf[(cb + c) * 32 + lane];  // b64, coalesced
            #pragma unroll
            for (int c = 0; c < CH; ++c)
                acc = __builtin_amdgcn_wmma_f32_16x16x4_f32(
                    false, aReg[cb + c], false, bReg[c], (short)0, acc, false, false);
        }
        #pragma unroll
        for (int e = 0; e < 8; ++e) {
            accMax[e] = fmaxf(accMax[e], acc[e]);
            accMin[e] = fminf(accMin[e], acc[e]);
        }
    }

    // C/D layout: VGPR e holds row (e + 8*(lane>>4)), col = lane&15.
    const int oIdx = oBase + (lane & 15);
    const float inv  = g[oIdx] / sqrtf(va[oIdx] + BN_EPS);
    const float mean = mu[oIdx];
    const float beta = be[oIdx];
    const int rowOff = (lane >> 4) << 3;
    #pragma unroll
    for (int e = 0; e < 8; ++e) {
        const int pp = ppBase + rowOff + e;
        float a1 = (accMax[e] - mean) * inv + beta;
        float a2 = (accMin[e] - mean) * inv + beta;
        a1 = LRELU(a1); a2 = LRELU(a2);
        out[(size_t)oIdx * outLD + outOff + pp] = fmaxf(a1, a2);
    }
}

// ---------------------------------------------------------------------------
// Global pooling over p: z[o] = max, z[O+o] = mean. One block per o.
// ---------------------------------------------------------------------------
__global__ void pool_kernel(const float* __restrict__ h5, float* __restrict__ z, int Np) {
    __shared__ float smax[256];
    __shared__ float ssum[256];
    const int o = blockIdx.x;
    const int t = threadIdx.x;
    const float* row = h5 + (size_t)o * Np;
    float mv = -INFINITY, sv = 0.f;
    for (int p = t; p < Np; p += 256) { float v = row[p]; mv = fmaxf(mv, v); sv += v; }
    smax[t] = mv; ssum[t] = sv;
    __syncthreads();
    for (int s = 128; s > 0; s >>= 1) {
        if (t < s) { smax[t] = fmaxf(smax[t], smax[t + s]); ssum[t] += ssum[t + s]; }
        __syncthreads();
    }
    if (t == 0) { z[o] = smax[0]; z[gridDim.x + o] = ssum[0] / (float)Np; }
}

// ---------------------------------------------------------------------------
// MLP head: 1024->512 (lrelu) -> 256 (lrelu, +bias) -> 11 (+bias) -> softmax.
// ---------------------------------------------------------------------------
__global__ void head_kernel(const float* __restrict__ z,
                            const float* __restrict__ lw1, const float* __restrict__ lw2,
                            const float* __restrict__ lb2, const float* __restrict__ ow,
                            const float* __restrict__ ob, float* __restrict__ out) {
    __shared__ float z1[512];
    __shared__ float z2[256];
    __shared__ float lg[11];
    const int t = threadIdx.x;  // 256 threads
    for (int o = t; o < 512; o += 256) {
        float s = 0.f;
        for (int j = 0; j < 1024; ++j) s += lw1[o * 1024 + j] * z[j];
        z1[o] = LRELU(s);
    }
    __syncthreads();
    {
        float s = lb2[t];
        for (int j = 0; j < 512; ++j) s += lw2[t * 512 + j] * z1[j];
        z2[t] = LRELU(s);
    }
    __syncthreads();
    if (t < 11) {
        float s = ob[t];
        for (int j = 0; j < 256; ++j) s += ow[t * 256 + j] * z2[j];
        lg[t] = s;
    }
    __syncthreads();
    if (t == 0) {
        float m = lg[0];
        for (int i = 1; i < 11; ++i) m = fmaxf(m, lg[i]);
        float e[11], sum = 0.f;
        for (int i = 0; i < 11; ++i) { e[i] = expf(lg[i] - m); sum += e[i]; }
        for (int i = 0; i < 11; ++i) out[i] = e[i] / sum;
    }
}

// ---------------------------------------------------------------------------
extern "C" void kernel_launch(void* const* d_in, const int* in_sizes, int n_in,
                              void* d_out, int out_size, void* d_ws, size_t ws_size,
                              hipStream_t stream) {
    (void)in_sizes; (void)n_in; (void)out_size; (void)ws_size;
    const float* X  = (const float*)d_in[0];
    const float* W1 = (const float*)d_in[1];
    const float *G1 = (const float*)d_in[2], *B1 = (const float*)d_in[3],
                *M1 = (const float*)d_in[4], *V1 = (const float*)d_in[5];
    const float* W2 = (const float*)d_in[6];
    const float *G2 = (const float*)d_in[7], *B2 = (const float*)d_in[8],
                *M2 = (const float*)d_in[9], *V2 = (const float*)d_in[10];
    const float* W3 = (const float*)d_in[11];
    const float *G3 = (const float*)d_in[12], *B3 = (const float*)d_in[13],
                *M3 = (const float*)d_in[14], *V3 = (const float*)d_in[15];
    const float* W4 = (const float*)d_in[16];
    const float *G4 = (const float*)d_in[17], *B4 = (const float*)d_in[18],
                *M4 = (const float*)d_in[19], *V4 = (const float*)d_in[20];
    const float* W5 = (const float*)d_in[21];
    const float *G5 = (const float*)d_in[22], *B5 = (const float*)d_in[23],
                *M5 = (const float*)d_in[24], *V5 = (const float*)d_in[25];
    const float* LW1 = (const float*)d_in[26];
    const float* LW2 = (const float*)d_in[27];
    const float* LB2 = (const float*)d_in[28];
    const float* OW  = (const float*)d_in[29];
    const float* OB  = (const float*)d_in[30];

    char* ws = (char*)d_ws;
    int*   idx1 = (int*)(ws + 0);            // 4096*20 ints
    float* x1   = (float*)(ws + 327680);     // 64*4096
    int*   idxs = (int*)(ws + 1376256);      // 128*20 ints (reused per stage)
    float* weff = (float*)(ws + 1386496);    // up to 21*128*256 (reused)
    float* x2   = (float*)(ws + 4139008);    // 64*8192
    float* x3   = (float*)(ws + 6236160);    // 128*16384
    float* x4   = (float*)(ws + 14624768);   // 256*32768
    float* h5   = (float*)(ws + 48179200);   // 512*32768
    float* z    = (float*)(ws + 115288064);  // 1024

    const int N = 4096, K = 20;

    // ---- stage 1: knn over points + 6->64 conv ----
    knn_big_kernel<<<N, 256, 0, stream>>>(X, N, K, idx1);
    stage1_conv_kernel<<<(64 * N) / 256, 256, 0, stream>>>(X, W1, G1, B1, M1, V1, idx1, x1, N, K);

    // ---- stage 2: X=(4096,64), O=64 -> x2 (64 x 8192) ----
    {
        const int M = 4096, P = 64, O = 64;
        knn_small_kernel<<<P, P, 0, stream>>>(x1, M, P, K, idxs);
        const int tot = (K + 1) * P * O;
        weff_prep_kernel<<<(tot + 255) / 256, 256, 0, stream>>>(weff, W2, idxs, P, O, K);
        const int tiles = (M / 16) * (O / 16);
        gemm_maxk_bn_kernel<64, true><<<(tiles + 7) / 8, 256, 0, stream>>>(
            x1, P, 1, M, O, weff, K, x2, 2 * M, 0, G2, B2, M2, V2);
        gemm_maxk_bn_kernel<64, true><<<(tiles + 7) / 8, 256, 0, stream>>>(
            x1, P, 1, M, O, weff + (size_t)K * P * O, 1, x2, 2 * M, M, G2, B2, M2, V2);
    }

    // ---- stage 3: X=(8192,64), O=128 -> x3 (128 x 16384) ----
    {
        const int M = 8192, P = 64, O = 128;
        knn_small_kernel<<<P, P, 0, stream>>>(x2, M, P, K, idxs);
        const int tot = (K + 1) * P * O;
        weff_prep_kernel<<<(tot + 255) / 256, 256, 0, stream>>>(weff, W3, idxs, P, O, K);
        const int tiles = (M / 16) * (O / 16);
        gemm_maxk_bn_kernel<64, true><<<(tiles + 7) / 8, 256, 0, stream>>>(
            x2, P, 1, M, O, weff, K, x3, 2 * M, 0, G3, B3, M3, V3);
        gemm_maxk_bn_kernel<64, true><<<(tiles + 7) / 8, 256, 0, stream>>>(
            x2, P, 1, M, O, weff + (size_t)K * P * O, 1, x3, 2 * M, M, G3, B3, M3, V3);
    }

    // ---- stage 4: X=(16384,128), O=256 -> x4 (256 x 32768) ----
    {
        const int M = 16384, P = 128, O = 256;
        knn_small_kernel<<<P, P, 0, stream>>>(x3, M, P, K, idxs);
        const int tot = (K + 1) * P * O;
        weff_prep_kernel<<<(tot + 255) / 256, 256, 0, stream>>>(weff, W4, idxs, P, O, K);
        const int tiles = (M / 16) * (O / 16);
        gemm_maxk_bn_kernel<128, true><<<(tiles + 7) / 8, 256, 0, stream>>>(
            x3, P, 1, M, O, weff, K, x4, 2 * M, 0, G4, B4, M4, V4);
        gemm_maxk_bn_kernel<128, true><<<(tiles + 7) / 8, 256, 0, stream>>>(
            x3, P, 1, M, O, weff + (size_t)K * P * O, 1, x4, 2 * M, M, G4, B4, M4, V4);
    }

    // ---- stage 5: h5 = lrelu(bn(W5 @ x4)), GEMM over c=256 ----
    {
        const int C = 256, O = 512, Np = 32768;
        // K=0 -> plain transpose path in fragment layout
        weff_prep_kernel<<<(C * O + 255) / 256, 256, 0, stream>>>(weff, W5, idxs, C, O, 0);
        const int tiles = (Np / 16) * (O / 16);
        // A[p,c] = x4[c*Np + p]  -> rsA=1, csA=Np
        gemm_maxk_bn_kernel<256, false><<<(tiles + 7) / 8, 256, 0, stream>>>(
            x4, 1, Np, Np, O, weff, 1, h5, Np, 0, G5, B5, M5, V5);
        pool_kernel<<<O, 256, 0, stream>>>(h5, z, Np);
        head_kernel<<<1, 256, 0, stream>>>(z, LW1, LW2, LB2, OW, OB, (float*)d_out);
    }
}